// Attention_spd_50706383897336
// MI455X (gfx1250) — compile-verified
//
#include <hip/hip_runtime.h>
#include <math.h>

typedef float v2f __attribute__((ext_vector_type(2)));
typedef float v8f __attribute__((ext_vector_type(8)));
typedef int v4i_t __attribute__((vector_size(16)));  // matches builtin proto

#define B_  8
#define N_  1024
#define DIM_ 512
#define H_  8
#define DH_ 64
#define INNER_ 512
#define SCALE_ 0.125f   // DH^-0.5
#define EPS_ 1e-12f

#define AS1 __attribute__((address_space(1)))
#define AS3 __attribute__((address_space(3)))

#if defined(__has_builtin)
#if __has_builtin(__builtin_amdgcn_global_load_async_to_lds_b128)
#define USE_ASYNC_BUILTIN 1
#endif
#if __has_builtin(__builtin_amdgcn_s_wait_asynccnt)
#define USE_WAIT_BUILTIN 1
#endif
#endif

// async DMA: 16 bytes global -> LDS, tracked by ASYNCcnt (no VGPR round-trip)
static __device__ __forceinline__ void async_cp16(float* lds_dst,
                                                  const float* gsrc) {
#ifdef USE_ASYNC_BUILTIN
  __builtin_amdgcn_global_load_async_to_lds_b128(
      (AS1 v4i_t*)(AS1 void*)const_cast<float*>(gsrc),
      (AS3 v4i_t*)(AS3 void*)lds_dst, 0, 0);
#else
  unsigned l = (unsigned)(unsigned long long)(AS3 char*)lds_dst;
  asm volatile("global_load_async_to_lds_b128 %0, %1, off" ::"v"(l), "v"(gsrc)
               : "memory");
#endif
}

template <int N>
static __device__ __forceinline__ void wait_async() {
#ifdef USE_WAIT_BUILTIN
  __builtin_amdgcn_s_wait_asynccnt(N);
#else
  asm volatile("s_wait_asynccnt %0" ::"n"(N) : "memory");
#endif
}

static __device__ __forceinline__ v8f wmma4(v2f a, v2f b, v8f c) {
  // D = A(16x4 f32) * B(4x16 f32) + C(16x16 f32)
  return __builtin_amdgcn_wmma_f32_16x16x4_f32(false, a, false, b, (short)0, c,
                                               false, false);
}

// ---------------------------------------------------------------------------
// fp32 WMMA GEMM, 64x64 tile / 256 threads (8 waves), async double-buffered.
// mode 0: C = A[8192,512] @ Wqkv[512,1536], scatter into q/k/v [B,H,N,DH]
// mode 1: C = A[8192,512] @ Wout[512,512] + bias, dense out [8192,512]
// ---------------------------------------------------------------------------
__global__ __launch_bounds__(256) void gemm_wmma_f32(
    const float* __restrict__ A, const float* __restrict__ Bm,
    const float* __restrict__ bias, float* __restrict__ qw,
    float* __restrict__ kw, float* __restrict__ vw, float* __restrict__ outp,
    int Kdim, int ldb, int mode) {
  __shared__ float As[2][64 * 20];  // stride 20 -> conflict-free A frags
  __shared__ float Bs[2][16 * 68];  // stride 68 -> conflict-free B frags

  const int tid = threadIdx.x;
  const int lane = tid & 31;
  const int w = tid >> 5;
  const int wm = w & 3;
  const int wn = w >> 2;
  const int m = lane & 15;
  const int khalf = lane >> 4;

  const int rowBase = blockIdx.y * 64;
  const int colBase = blockIdx.x * 64;

  const int ar = tid >> 2, ac = (tid & 3) * 4;
  const int br = tid >> 4, bc = (tid & 15) * 4;

  v8f acc0 = {0, 0, 0, 0, 0, 0, 0, 0};
  v8f acc1 = {0, 0, 0, 0, 0, 0, 0, 0};

  // prologue: stage tile 0 (2 async b128 per thread per stage)
  async_cp16(&As[0][ar * 20 + ac], A + (size_t)(rowBase + ar) * Kdim + ac);
  async_cp16(&Bs[0][br * 68 + bc], Bm + (size_t)br * ldb + colBase + bc);

  const int nstage = Kdim >> 4;
  for (int s = 0; s < nstage; ++s) {
    const int cur = s & 1;
    if (s + 1 < nstage) {
      int k0 = (s + 1) * 16;
      async_cp16(&As[cur ^ 1][ar * 20 + ac],
                 A + (size_t)(rowBase + ar) * Kdim + k0 + ac);
      async_cp16(&Bs[cur ^ 1][br * 68 + bc],
                 Bm + (size_t)(k0 + br) * ldb + colBase + bc);
      wait_async<2>();  // current tile landed; next still in flight
    } else {
      wait_async<0>();
    }
    __syncthreads();
#pragma unroll
    for (int kk = 0; kk < 16; kk += 4) {
      int kl = kk + 2 * khalf;
      v2f a;
      a.x = As[cur][(wm * 16 + m) * 20 + kl];
      a.y = As[cur][(wm * 16 + m) * 20 + kl + 1];
      int bn = wn * 32 + m;
      v2f b0, b1;
      b0.x = Bs[cur][kl * 68 + bn];
      b0.y = Bs[cur][(kl + 1) * 68 + bn];
      b1.x = Bs[cur][kl * 68 + bn + 16];
      b1.y = Bs[cur][(kl + 1) * 68 + bn + 16];
      acc0 = wmma4(a, b0, acc0);
      acc1 = wmma4(a, b1, acc1);
    }
    __syncthreads();
  }

  // epilogue: vgpr r, lanes0-15: M=r, lanes16-31: M=8+r, N=lane&15
#pragma unroll
  for (int r = 0; r < 8; ++r) {
    int row = rowBase + wm * 16 + r + 8 * khalf;
#pragma unroll
    for (int a = 0; a < 2; ++a) {
      int col = colBase + wn * 32 + a * 16 + m;
      float val = (a == 0) ? acc0[r] : acc1[r];
      if (mode == 1) {
        outp[(size_t)row * INNER_ + col] = val + bias[col];
      } else {
        int which = col >> 9;  // 0=q 1=k 2=v
        int inner = col & 511;
        int h = inner >> 6, d = inner & 63;
        int b = row >> 10, n = row & 1023;
        float* dst = (which == 0) ? qw : (which == 1) ? kw : vw;
        dst[(((size_t)b * H_ + h) * N_ + n) * DH_ + d] = val;
      }
    }
  }
}

// ---------------------------------------------------------------------------
// Fused attention: one WG (128 thr / 4 waves) per (b,h, 16-row block).
// dots[16,1024] in LDS; K/V tiles async double-buffered; spd re-read from L2.
// LDS = 105,344 B -> 2 workgroups per WGP.
// ---------------------------------------------------------------------------
__global__ __launch_bounds__(128) void attn_fused(
    const float* __restrict__ q, const float* __restrict__ k,
    const float* __restrict__ v, const float* __restrict__ spd,
    const float* __restrict__ head_mask, float* __restrict__ aout) {
  extern __shared__ float smem[];
  float* dots = smem;          // 16*1028 = 16448
  float* stg0 = smem + 16448;  // 64*68  = 4352
  float* stg1 = smem + 20800;  // 64*68  = 4352
  float* Qs = smem + 25152;    // 16*68  = 1088
  float* red = smem + 26240;   // 96 scalars  (total 26336 floats)

  const int rb = blockIdx.x;  // 0..63 row block
  const int bh = blockIdx.y;  // 0..63
  const int b = bh >> 3, h = bh & 7;
  const int r0 = rb * 16;

  const float* qb = q + (size_t)bh * N_ * DH_;
  const float* kb = k + (size_t)bh * N_ * DH_;
  const float* vb = v + (size_t)bh * N_ * DH_;

  const int tid = threadIdx.x;
  const int lane = tid & 31;
  const int w = tid >> 5;
  const int m = lane & 15;
  const int khalf = lane >> 4;

  // head-dropout rescale factor
  float msum = 0.f;
#pragma unroll
  for (int i = 0; i < H_; ++i) msum += head_mask[i];
  const float sm = head_mask[h] * (float)H_ / msum;

  // ---- async prologue: Q block + K tile 0 ----
  for (int i = tid; i < 256; i += 128) {
    int r = i >> 4, c = (i & 15) * 4;
    async_cp16(Qs + r * 68 + c, qb + (size_t)(r0 + r) * DH_ + c);
  }
  for (int i = tid; i < 1024; i += 128) {
    int r = i >> 4, c = (i & 15) * 4;
    async_cp16(stg0 + r * 68 + c, kb + (size_t)r * DH_ + c);
  }
  wait_async<8>();  // Q landed (K0 may still be in flight)
  __syncthreads();

  // Q fragments depend only on kk: hoist all 16 into registers
  v2f afr[16];
#pragma unroll
  for (int kk = 0; kk < 64; kk += 4) {
    int kl = kk + 2 * khalf;
    afr[kk >> 2].x = Qs[m * 68 + kl];
    afr[kk >> 2].y = Qs[m * 68 + kl + 1];
  }

  // ---- dots = (Q Kt) * SCALE, double-buffered async K tiles ----
  for (int it = 0; it < 16; ++it) {
    const int cur = it & 1;
    float* nb = cur ? stg0 : stg1;
    const float* src = (it < 15) ? (kb + (size_t)(it + 1) * 64 * DH_) : vb;
    for (int i = tid; i < 1024; i += 128) {  // prefetch next K (or V tile 0)
      int r = i >> 4, c = (i & 15) * 4;
      async_cp16(nb + r * 68 + c, src + (size_t)r * DH_ + c);
    }
    wait_async<8>();  // current tile landed
    __syncthreads();
    float* sb = cur ? stg1 : stg0;
    int nt = it * 4 + w;
    v8f acc = {0, 0, 0, 0, 0, 0, 0, 0};
#pragma unroll
    for (int kk = 0; kk < 64; kk += 4) {
      int kl = kk + 2 * khalf;
      int nrow = w * 16 + m;
      v2f bf;
      bf.x = sb[nrow * 68 + kl];
      bf.y = sb[nrow * 68 + kl + 1];
      acc = wmma4(afr[kk >> 2], bf, acc);
    }
#pragma unroll
    for (int r = 0; r < 8; ++r) {
      int mm = r + 8 * khalf;
      dots[mm * 1028 + nt * 16 + m] = acc[r] * SCALE_;
    }
    __syncthreads();
  }
  // note: V tile 0 is now streaming into stg0 under the softmax passes

  const float* spdb = spd + ((size_t)b * N_ + r0) * N_;

  // ---- pass1: row norms of dots and prod = dots*spd ----
  for (int rr = 0; rr < 4; ++rr) {
    int r = w * 4 + rr;
    float sp2 = 0.f, sd2 = 0.f;
    for (int j = 0; j < 32; ++j) {
      int c = j * 32 + lane;
      float d = dots[r * 1028 + c];
      float p = d * spdb[(size_t)r * N_ + c];
      sp2 += p * p;
      sd2 += d * d;
    }
    for (int off = 16; off; off >>= 1) {
      sp2 += __shfl_xor(sp2, off);
      sd2 += __shfl_xor(sd2, off);
    }
    if (lane == 0) { red[r] = sp2; red[16 + r] = sd2; }
  }
  __syncthreads();
  if (tid < 16)
    red[32 + tid] = sqrtf(red[16 + tid]) / fmaxf(sqrtf(red[tid]), EPS_);
  __syncthreads();

  // ---- pass2: row max of logits = dots*(1 + spd*f) ----
  for (int rr = 0; rr < 4; ++rr) {
    int r = w * 4 + rr;
    float f = red[32 + r];
    float mx = -INFINITY;
    for (int j = 0; j < 32; ++j) {
      int c = j * 32 + lane;
      float d = dots[r * 1028 + c];
      mx = fmaxf(mx, d + d * spdb[(size_t)r * N_ + c] * f);
    }
    for (int off = 16; off; off >>= 1) mx = fmaxf(mx, __shfl_xor(mx, off));
    if (lane == 0) red[48 + r] = mx;
  }
  __syncthreads();

  // ---- pass3: dots <- exp(logit - max), row sums ----
  for (int rr = 0; rr < 4; ++rr) {
    int r = w * 4 + rr;
    float f = red[32 + r], mx = red[48 + r];
    float s = 0.f;
    for (int j = 0; j < 32; ++j) {
      int c = j * 32 + lane;
      float d = dots[r * 1028 + c];
      float e = __expf(d + d * spdb[(size_t)r * N_ + c] * f - mx);
      dots[r * 1028 + c] = e;
      s += e;
    }
    for (int off = 16; off; off >>= 1) s += __shfl_xor(s, off);
    if (lane == 0) red[64 + r] = s;
  }
  __syncthreads();
  if (tid < 16) red[80 + tid] = sm / red[64 + tid];  // folded into epilogue
  __syncthreads();

  // ---- out = attn @ V, double-buffered async V tiles (V0 already issued) ----
  v8f oacc = {0, 0, 0, 0, 0, 0, 0, 0};
  const int d0 = w * 16;
  for (int it = 0; it < 16; ++it) {
    const int cur = it & 1;
    if (it < 15) {
      float* nb = cur ? stg0 : stg1;
      const float* src = vb + (size_t)(it + 1) * 64 * DH_;
      for (int i = tid; i < 1024; i += 128) {
        int r = i >> 4, c = (i & 15) * 4;
        async_cp16(nb + r * 68 + c, src + (size_t)r * DH_ + c);
      }
      wait_async<8>();
    } else {
      wait_async<0>();
    }
    __syncthreads();
    float* sb = cur ? stg1 : stg0;
#pragma unroll
    for (int kk = 0; kk < 64; kk += 4) {
      int kl = kk + 2 * khalf;
      v2f a, bf;
      a.x = dots[m * 1028 + it * 64 + kl];
      a.y = dots[m * 1028 + it * 64 + kl + 1];
      bf.x = sb[kl * 68 + d0 + m];
      bf.y = sb[(kl + 1) * 68 + d0 + m];
      oacc = wmma4(a, bf, oacc);
    }
    __syncthreads();
  }

  float* ob = aout + ((size_t)b * N_ + r0) * INNER_ + h * DH_;
#pragma unroll
  for (int r = 0; r < 8; ++r) {
    int mm = r + 8 * khalf;
    ob[(size_t)mm * INNER_ + d0 + m] = oacc[r] * red[80 + mm];
  }
}

// ---------------------------------------------------------------------------
extern "C" void kernel_launch(void* const* d_in, const int* in_sizes, int n_in,
                              void* d_out, int out_size, void* d_ws,
                              size_t ws_size, hipStream_t stream) {
  const float* x = (const float*)d_in[0];          // [8,1024,512]
  const float* spd = (const float*)d_in[1];        // [8,1,1024,1024]
  const float* head_mask = (const float*)d_in[2];  // [8]
  const float* W_qkv = (const float*)d_in[3];      // [512,1536]
  const float* W_out = (const float*)d_in[4];      // [512,512]
  const float* b_out = (const float*)d_in[5];      // [512]
  float* out = (float*)d_out;                      // [8,1024,512]

  float* ws = (float*)d_ws;
  const size_t per = (size_t)B_ * H_ * N_ * DH_;  // 4,194,304 floats
  float* qw = ws;
  float* kw = ws + per;
  float* vw = ws + 2 * per;
  float* aout = ws + 3 * per;  // [B,N,INNER]

  // 1) QKV projection + head split
  {
    dim3 grid(3 * INNER_ / 64, (B_ * N_) / 64);  // (24, 128)
    gemm_wmma_f32<<<grid, 256, 0, stream>>>(x, W_qkv, nullptr, qw, kw, vw,
                                            nullptr, DIM_, 3 * INNER_, 0);
  }
  // 2) fused attention core
  {
    dim3 grid(N_ / 16, B_ * H_);  // (64, 64)
    size_t shmem = (size_t)26336 * sizeof(float);  // 105,344 B -> 2 WG/WGP
    attn_fused<<<grid, 128, shmem, stream>>>(qw, kw, vw, spd, head_mask, aout);
  }
  // 3) output projection + bias
  {
    dim3 grid(DIM_ / 64, (B_ * N_) / 64);  // (8, 128)
    gemm_wmma_f32<<<grid, 256, 0, stream>>>(aout, W_out, b_out, nullptr,
                                            nullptr, nullptr, out, INNER_,
                                            DIM_, 1);
  }
}